// GCN_65618510348474
// MI455X (gfx1250) — compile-verified
//
#include <hip/hip_runtime.h>
#include <hip/hip_bf16.h>

typedef __attribute__((ext_vector_type(2))) float v2f;
typedef __attribute__((ext_vector_type(8))) float v8f;

#define D 128  // D_IN == D_OUT == 128

// ---------------------------------------------------------------- degree ----
__global__ void init_deg_kernel(float* deg, int n) {
  int i = blockIdx.x * blockDim.x + threadIdx.x;
  if (i < n) deg[i] = 1.0f;  // self-loop weight
}

__global__ void deg_accum_kernel(const int* __restrict__ col,
                                 const float* __restrict__ ew,
                                 float* deg, int e_cnt) {
  int e = blockIdx.x * blockDim.x + threadIdx.x;
  if (e < e_cnt) atomicAdd(&deg[col[e]], ew[e]);
}

__global__ void dinv_kernel(float* deg, int n) {
  int i = blockIdx.x * blockDim.x + threadIdx.x;
  if (i < n) {
    float d = deg[i];
    deg[i] = (d > 0.0f) ? rsqrtf(d) : 0.0f;  // in-place: deg becomes dinv
  }
}

// ------------------------------------------------------------- WMMA GEMM ----
// xw[N x 128] = x[N x 128] @ w[128 x 128], fp32 via V_WMMA_F32_16X16X4_F32.
// One wave32 computes a 16-row x 128-col tile; K loop in steps of 4.
// A (16x4 f32) layout: lanes 0-15 -> M=lane, V0=K+0,V1=K+1; lanes 16-31 -> K+2,K+3.
// B (4x16 f32) layout: V0: lanes 0-15 = row K+0 (N=lane), lanes 16-31 = row K+2.
// C/D (16x16 f32): VGPR j: lanes 0-15 -> M=j, N=lane; lanes 16-31 -> M=8+j.
__global__ __launch_bounds__(256) void gemm_wmma_kernel(
    const float* __restrict__ x, const float* __restrict__ w,
    float* __restrict__ xw, int n_rows) {
  const int lane = threadIdx.x & 31;
  const int wave = blockIdx.x * (blockDim.x >> 5) + (threadIdx.x >> 5);
  const int row0 = wave * 16;
  if (row0 >= n_rows) return;  // whole-wave exit keeps EXEC all-1s for WMMA

  const int m  = lane & 15;        // row within tile (lanes 0-15 / 16-31)
  const int kh = (lane >> 4) * 2;  // K sub-offset: 0 or 2

  // clamp so all 32 lanes stay active even on a ragged last tile
  const int arow = (row0 + m < n_rows) ? (row0 + m) : (n_rows - 1);

  v8f acc[8];
#pragma unroll
  for (int t = 0; t < 8; ++t) acc[t] = (v8f){0.f,0.f,0.f,0.f,0.f,0.f,0.f,0.f};

  for (int k = 0; k < D; k += 4) {
    const float* ap = x + (size_t)arow * D + (k + kh);
    v2f a;
    a.x = ap[0];
    a.y = ap[1];
#pragma unroll
    for (int t = 0; t < 8; ++t) {
      const float* bp = w + (size_t)(k + kh) * D + t * 16 + m;
      v2f b;
      b.x = bp[0];      // row k+kh
      b.y = bp[D];      // row k+kh+1
      acc[t] = __builtin_amdgcn_wmma_f32_16x16x4_f32(
          /*neg_a=*/false, a, /*neg_b=*/false, b,
          /*c_mod=*/(short)0, acc[t], /*reuse_a=*/false, /*reuse_b=*/false);
    }
  }

#pragma unroll
  for (int t = 0; t < 8; ++t) {
#pragma unroll
    for (int j = 0; j < 8; ++j) {
      const int M = (kh << 2) + j;  // kh=0 -> rows 0..7, kh=2 -> rows 8..15
      if (row0 + M < n_rows)
        xw[(size_t)(row0 + M) * D + t * 16 + m] = acc[t][j];
    }
  }
}

// -------------------------------------------------- self-loop + bias init ---
__global__ void init_out_kernel(const float* __restrict__ xw,
                                const float* __restrict__ dinv,
                                const float* __restrict__ bias,
                                float* out, long long total) {
  long long idx = (long long)blockIdx.x * blockDim.x + threadIdx.x;
  if (idx >= total) return;
  int i = (int)(idx >> 7);       // node
  int d = (int)(idx & (D - 1));  // feature
  float di = dinv[i];
  out[idx] = bias[d] + di * di * xw[idx];  // self-loop: norm = dinv*1*dinv
}

// ------------------------------------------------------------ edge scatter --
__global__ void scatter_kernel(const int* __restrict__ row,
                               const int* __restrict__ col,
                               const float* __restrict__ ew,
                               const float* __restrict__ dinv,
                               const float* __restrict__ xw,
                               float* out, long long total) {
  long long idx = (long long)blockIdx.x * blockDim.x + threadIdx.x;
  if (idx >= total) return;
  int e = (int)(idx >> 7);
  int d = (int)(idx & (D - 1));
  int r = row[e];
  int c = col[e];
  float norm = dinv[r] * ew[e] * dinv[c];
  atomicAdd(&out[(size_t)c * D + d], norm * xw[(size_t)r * D + d]);
}

// ----------------------------------------------------------------- PReLU ----
__global__ void prelu_kernel(float* out, const float* __restrict__ pa,
                             long long total) {
  long long idx = (long long)blockIdx.x * blockDim.x + threadIdx.x;
  if (idx >= total) return;
  float v = out[idx];
  out[idx] = (v >= 0.0f) ? v : pa[0] * v;
}

// ---------------------------------------------------------------------------
extern "C" void kernel_launch(void* const* d_in, const int* in_sizes, int n_in,
                              void* d_out, int out_size, void* d_ws, size_t ws_size,
                              hipStream_t stream) {
  const float* x    = (const float*)d_in[0];  // [N, 128]
  const int*   ei   = (const int*)  d_in[1];  // [2, E] flattened (rows, cols)
  const float* ew   = (const float*)d_in[2];  // [E]
  const float* w    = (const float*)d_in[3];  // [128, 128]
  const float* bias = (const float*)d_in[4];  // [128]
  const float* pa   = (const float*)d_in[5];  // [1]
  float* out = (float*)d_out;                 // [N, 128]

  const int n     = in_sizes[0] / D;
  const int e_cnt = in_sizes[2];
  const int* row = ei;          // sources
  const int* col = ei + e_cnt;  // targets

  // workspace: deg/dinv [n] floats, then xw [n*D] floats (~51.6 MB total)
  float* deg = (float*)d_ws;
  float* xw  = deg + n;

  const long long nd  = (long long)n * D;
  const long long edd = (long long)e_cnt * D;

  // 1) deg = 1 (self-loops)
  init_deg_kernel<<<(n + 255) / 256, 256, 0, stream>>>(deg, n);
  // 2) deg[col] += ew
  deg_accum_kernel<<<(e_cnt + 255) / 256, 256, 0, stream>>>(col, ew, deg, e_cnt);
  // 3) dinv = rsqrt(deg) in-place
  dinv_kernel<<<(n + 255) / 256, 256, 0, stream>>>(deg, n);
  // 4) xw = x @ w via WMMA; one wave per 16-row tile, 8 waves per block
  {
    int tiles  = (n + 15) / 16;
    int blocks = (tiles + 7) / 8;
    gemm_wmma_kernel<<<blocks, 256, 0, stream>>>(x, w, xw, n);
  }
  // 5) out = bias + dinv^2 * xw
  init_out_kernel<<<(int)((nd + 255) / 256), 256, 0, stream>>>(xw, deg, bias, out, nd);
  // 6) out[col] += dinv[row]*ew*dinv[col] * xw[row]
  scatter_kernel<<<(int)((edd + 255) / 256), 256, 0, stream>>>(row, col, ew, deg, xw, out, edd);
  // 7) PReLU
  prelu_kernel<<<(int)((nd + 255) / 256), 256, 0, stream>>>(out, pa, nd);
}